// HJBLoss_68925635166304
// MI455X (gfx1250) — compile-verified
//
#include <hip/hip_runtime.h>

typedef __attribute__((ext_vector_type(2))) float v2f;
typedef __attribute__((ext_vector_type(8))) float v8f;

// Exact-f32 full-wave (32-lane) sum using one V_WMMA_F32_16X16X4_F32.
// A per-lane = {a, 0}  ->  A[m][0] = a_m, A[m][2] = a_{m+16}, other cols 0.
// B = all-ones 4x16    ->  D[m][n] = a_m + a_{m+16}  (C = 0).
// C/D layout: lane<16 VGPR j = D[j][lane]; lane>=16 VGPR j = D[j+8][lane-16].
// Sum of 8 C components: lanes 0-15 -> S_lo, lanes 16-31 -> S_hi.
// Total = S_lo + S_hi via one shfl_xor(16).
// Requires EXEC all-ones (call only from fully-converged full waves).
__device__ __forceinline__ float wave_sum_wmma(float a) {
    v2f A; A[0] = a;    A[1] = 0.0f;
    v2f Bm; Bm[0] = 1.0f; Bm[1] = 1.0f;
    v8f C = {};
    C = __builtin_amdgcn_wmma_f32_16x16x4_f32(
        /*neg_a=*/false, A, /*neg_b=*/false, Bm,
        /*c_mod=*/(short)0, C, /*reuse_a=*/false, /*reuse_b=*/false);
    float h = ((C[0] + C[1]) + (C[2] + C[3])) + ((C[4] + C[5]) + (C[6] + C[7]));
    h += __shfl_xor(h, 16, 32);   // S_lo + S_hi on every lane
    return h;
}

// Stage 1: stream all rows, one f32 partial per block.
__global__ void __launch_bounds__(256)
hjb_main_kernel(const float* __restrict__ X,    // [B,4]
                const float* __restrict__ mu,   // [B,2]
                const float* __restrict__ sig,  // [B]
                const float* __restrict__ u,    // [B,2]
                const float* __restrict__ Q,    // [4,4]
                const float* __restrict__ R,    // [2,2]
                const float* __restrict__ xt,   // [4]
                float* __restrict__ partials,   // [gridDim.x]
                int B)
{
    const int tid    = blockIdx.x * blockDim.x + threadIdx.x;
    const int stride = gridDim.x * blockDim.x;

    // Uniform small constants (scalar-loaded; hot in constant/L2 cache).
    const float q00 = Q[0],  q01 = Q[1],  q02 = Q[2],  q03 = Q[3];
    const float q10 = Q[4],  q11 = Q[5],  q12 = Q[6],  q13 = Q[7];
    const float q20 = Q[8],  q21 = Q[9],  q22 = Q[10], q23 = Q[11];
    const float q30 = Q[12], q31 = Q[13], q32 = Q[14], q33 = Q[15];
    const float r00 = R[0],  r01 = R[1],  r10 = R[2],  r11 = R[3];
    const float xt0 = xt[0], xt1 = xt[1], xt2 = xt[2], xt3 = xt[3];
    // trace(2Q * COV*COV^T), COV*COV^T = diag(0,0,0.25,0.25)
    const float tr_const = 0.5f * (q22 + q33);

    const float4* __restrict__ Xv = (const float4*)X;
    const float2* __restrict__ Uv = (const float2*)u;
    const float2* __restrict__ Mv = (const float2*)mu;

    float acc = 0.0f;
    #pragma unroll 4
    for (int r = tid; r < B; r += stride) {
        const float4 x  = Xv[r];
        const float2 uu = Uv[r];
        const float2 m  = Mv[r];
        const float  s  = sig[r];

        const float xr0 = x.x - xt0, xr1 = x.y - xt1;
        const float xr2 = x.z - xt2, xr3 = x.w - xt3;

        // t = Q^T xr  (row vector xr^T Q)
        const float t0 = xr0*q00 + xr1*q10 + xr2*q20 + xr3*q30;
        const float t1 = xr0*q01 + xr1*q11 + xr2*q21 + xr3*q31;
        const float t2 = xr0*q02 + xr1*q12 + xr2*q22 + xr3*q32;
        const float t3 = xr0*q03 + xr1*q13 + xr2*q23 + xr3*q33;

        const float V = t0*xr0 + t1*xr1 + t2*xr2 + t3*xr3;

        // dyn = f X + G u + COV mu  (f, G, COV hard-coded, omega = 0.6)
        const float dyn0 =  x.z + 0.30f*uu.x;
        const float dyn1 =  x.w + 0.25f*uu.y;
        const float dyn2 =  0.6f*x.y + uu.x + 0.5f*m.x;
        const float dyn3 = -0.6f*x.x + uu.y + 0.5f*m.y;

        const float g1dyn = 2.0f*(t0*dyn0 + t1*dyn1 + t2*dyn2 + t3*dyn3);
        const float uRu   = 0.5f*(uu.x*(r00*uu.x + r01*uu.y) +
                                  uu.y*(r10*uu.x + r11*uu.y));
        const float trterm = 0.5f * s * s * tr_const;

        acc += g1dyn + V + uRu + trterm;
    }

    // Wave reduction via WMMA (exact f32), then cross-wave via LDS.
    const float wsum = wave_sum_wmma(acc);

    __shared__ float lds[8];                 // 256 threads = 8 wave32
    const int lane = threadIdx.x & 31;
    const int wid  = threadIdx.x >> 5;
    if (lane == 0) lds[wid] = wsum;
    __syncthreads();
    if (threadIdx.x == 0) {
        float b = 0.0f;
        #pragma unroll
        for (int i = 0; i < 8; ++i) b += lds[i];
        partials[blockIdx.x] = b;
    }
}

// Stage 2: one wave folds the block partials and writes the mean.
__global__ void __launch_bounds__(32)
hjb_reduce_kernel(const float* __restrict__ partials, int n,
                  float inv_b, float* __restrict__ out)
{
    float a = 0.0f;
    for (int i = threadIdx.x; i < n; i += 32) a += partials[i];
    const float total = wave_sum_wmma(a);
    if (threadIdx.x == 0) out[0] = total * inv_b;
}

extern "C" void kernel_launch(void* const* d_in, const int* in_sizes, int n_in,
                              void* d_out, int out_size, void* d_ws, size_t ws_size,
                              hipStream_t stream)
{
    const float* X   = (const float*)d_in[0];
    const float* mu  = (const float*)d_in[1];
    const float* sig = (const float*)d_in[2];
    const float* u   = (const float*)d_in[3];
    const float* Q   = (const float*)d_in[4];
    const float* R   = (const float*)d_in[5];
    const float* xt  = (const float*)d_in[6];
    float* out = (float*)d_out;

    const int B = in_sizes[2];               // sigma has exactly B elements

    const int NT = 256;
    const int NB = 1024;                     // 262,144 threads -> 16 rows/thread
    float* partials = (float*)d_ws;          // 4 KB of workspace

    hjb_main_kernel<<<NB, NT, 0, stream>>>(X, mu, sig, u, Q, R, xt, partials, B);
    hjb_reduce_kernel<<<1, 32, 0, stream>>>(partials, NB, 1.0f / (float)B, out);
}